// MyModel_50672024158788
// MI455X (gfx1250) — compile-verified
//
#include <hip/hip_runtime.h>

#define T_STEPS 50000
#define HID     512
#define CHUNK   1000                 // timesteps staged per TDM transfer
#define NCHUNK  (T_STEPS / CHUNK)    // 50, divides evenly
#define CH_FLOATS (CHUNK * 3)        // 3000 dwords per chunk
#define CH_BYTES  (CH_FLOATS * 4)    // 12000 bytes

typedef __attribute__((ext_vector_type(4))) unsigned int u32x4;
typedef __attribute__((ext_vector_type(8))) int          i32x8;
typedef __attribute__((ext_vector_type(4))) int          i32x4;

// Issue one Tensor-Data-Mover load: CH_BYTES contiguous bytes of global -> LDS.
// D# built per CDNA5 ISA:
//  group0: [1:0]=count=1, [63:32]=lds_addr, [120:64]=global_addr, [127:126]=type=2
//  group1: data_size=4B (code 2); 1-D tile: tile_dim0 = tensor_dim0 = stride = CH_FLOATS
__device__ __forceinline__ void tdm_load_chunk(unsigned lds_addr,
                                               unsigned long long gaddr) {
  u32x4 g0;
  g0.x = 1u;                                            // count=1, user descriptor
  g0.y = lds_addr;                                      // LDS byte address
  g0.z = (unsigned)(gaddr & 0xFFFFFFFFull);             // global_addr[31:0]
  g0.w = ((unsigned)(gaddr >> 32) & 0x01FFFFFFu)        // global_addr[56:32]
         | 0x80000000u;                                 // type = 2 ("image")
  i32x8 g1;
  g1[0] = (2 << 16);                   // data_size=4B; mask/flags/pad = 0
  g1[1] = (int)((CH_FLOATS & 0xFFFF) << 16);   // tensor_dim0[15:0] @ bits[63:48]
  g1[2] = (int)(CH_FLOATS >> 16);              // tensor_dim0[31:16]; tensor_dim1=0
  g1[3] = (int)((unsigned)CH_FLOATS << 16);    // tile_dim0 @ bits[127:112]
  g1[4] = 0;                                   // tile_dim1=0 (1-D), tile_dim2=0
  g1[5] = CH_FLOATS;                           // tensor_dim0_stride[31:0]
  g1[6] = 0;
  g1[7] = 0;
  i32x4 gz = {0, 0, 0, 0};                     // groups 2/3 unused (dims 2..4 = 0)
  i32x8 z8 = {0, 0, 0, 0, 0, 0, 0, 0};         // trailing group (6-arg form), unused
  __builtin_amdgcn_tensor_load_to_lds(g0, g1, gz, gz, z8, 0);
}

__global__ __launch_bounds__(32)
void hydro_scan_kernel(const float* __restrict__ x,
                       const float* __restrict__ f_,
                       const float* __restrict__ smax_,
                       const float* __restrict__ qmax_,
                       const float* __restrict__ df_,
                       const float* __restrict__ tmax_,
                       const float* __restrict__ tmin_,
                       float* __restrict__ q) {
  __shared__ float sbuf[2][CH_FLOATS];         // 24 KB double buffer

  const int h = blockIdx.x * 32 + threadIdx.x; // one hidden unit per lane

  // Per-unit parameters -> registers, plus hoisted invariants.
  const float f    = f_[h];
  const float smax = smax_[h];
  const float qmax = qmax_[h];
  const float df   = df_[h];
  const float tmax = tmax_[h];
  const float tmin = tmin_[h];

  const float inv_smax = 1.0f / smax;
  const float fl2e = f * 1.44269504088896340736f;   // f * log2(e)
  const float carg = -fl2e * smax;                  // exp arg = fl2e*s1 + carg

  const unsigned long long xg = (unsigned long long)(size_t)x;
  const unsigned lds0 = (unsigned)(unsigned long long)(size_t)(&sbuf[0][0]);
  const unsigned lds1 = (unsigned)(unsigned long long)(size_t)(&sbuf[1][0]);

  // Prime the pipeline: stage chunk 0.
  tdm_load_chunk(lds0, xg);

  float s0 = 0.0f, s1 = 0.0f;
  float* qp = q + h;

  for (int c = 0; c < NCHUNK; ++c) {
    const float* __restrict__ buf = &sbuf[c & 1][0];
    if (c + 1 < NCHUNK) {
      // Kick off next chunk into the other buffer, then wait for chunk c
      // (TDM transfers complete in issue order; <=1 outstanding keeps c done).
      tdm_load_chunk((c & 1) ? lds0 : lds1,
                     xg + (unsigned long long)(c + 1) * CH_BYTES);
      __builtin_amdgcn_s_wait_tensorcnt(1);
    } else {
      __builtin_amdgcn_s_wait_tensorcnt(0);
    }

#pragma unroll 2
    for (int i = 0; i < CHUNK; ++i) {
      const float p  = buf[i * 3 + 0];
      const float T  = buf[i * 3 + 1];
      const float ld = buf[i * 3 + 2];

      // Hamon PET — depends only on inputs, entirely off the recurrence chain.
      const float esat = 0.611f * __builtin_amdgcn_exp2f(
          (1.44269504088896340736f * 17.3f) * T *
          __builtin_amdgcn_rcpf(T + 237.3f));
      const float pe = (29.8f * 24.0f) * ld * esat *
                       __builtin_amdgcn_rcpf(T + 273.2f);

      // Branch-free EXP-HYDRO step.
      const bool  snow = (T <= tmin);
      const float ps   = snow ? p : 0.0f;
      const float pr   = p - ps;
      const float melt = fminf(s0, fmaxf(df * (T - tmax), 0.0f)); // s0 >= 0
      const float et   = pe * fminf(s1 * inv_smax, 1.0f);
      const float qb   = fminf(qmax,
          qmax * __builtin_amdgcn_exp2f(__builtin_fmaf(fl2e, s1, carg)));
      const float qs   = fmaxf(s1 - smax, 0.0f);
      const float out  = qb + qs;

      s0 = fmaxf(s0 + ps - melt, 0.0f);
      s1 = fmaxf(((s1 + pr) + melt) - et - out, 0.0f);

      // Coalesced 128B/wave store; fire-and-forget into L2 (output fits in L2).
      qp[(size_t)(c * CHUNK + i) * HID] = out;
    }
  }
}

extern "C" void kernel_launch(void* const* d_in, const int* in_sizes, int n_in,
                              void* d_out, int out_size, void* d_ws,
                              size_t ws_size, hipStream_t stream) {
  (void)in_sizes; (void)n_in; (void)out_size; (void)d_ws; (void)ws_size;
  const float* x    = (const float*)d_in[0];
  const float* f    = (const float*)d_in[1];
  const float* smax = (const float*)d_in[2];
  const float* qmax = (const float*)d_in[3];
  const float* df   = (const float*)d_in[4];
  const float* tmax = (const float*)d_in[5];
  const float* tmin = (const float*)d_in[6];
  float* q = (float*)d_out;

  // 512 units / 32 lanes = 16 single-wave workgroups -> one serial chain per SIMD.
  hydro_scan_kernel<<<HID / 32, 32, 0, stream>>>(x, f, smax, qmax, df, tmax,
                                                 tmin, q);
}